// ScaledDotProductAttention_54984171324142
// MI455X (gfx1250) — compile-verified
//
#include <hip/hip_runtime.h>

typedef float v2f __attribute__((ext_vector_type(2)));
typedef float v8f __attribute__((ext_vector_type(8)));

#define BATCH   64
#define SEQ     8192
#define DIM     64
#define NCHUNK  32
#define CHUNK   256          // rows per block
#define WAVES   4
#define RPW     64           // rows per wave (CHUNK/WAVES)
#define TPW     4            // 16-row WMMA tiles per wave
#define WSSTR   68           // per-(b,chunk) ws floats: [0]=m, [1]=l, [2..65]=acc[64]

// ---------------------------------------------------------------------------
// Pass 1: one block = one (batch, 256-row chunk).
//   scores (q.K^T / 8) via v_wmma_f32_16x16x4_f32, chunk-local softmax
//   partials (m, l) + partial P.V accumulation, raw scores -> attn region.
// ---------------------------------------------------------------------------
__global__ __launch_bounds__(128) void attn_pass1(
    const float* __restrict__ q,
    const float* __restrict__ k,
    const float* __restrict__ v,
    float* __restrict__ attn_raw,   // d_out + 4096, raw scaled scores
    float* __restrict__ ws)
{
    __shared__ float s_q[DIM];        // q / temperature
    __shared__ float s_sc[CHUNK];     // raw scores for this chunk
    __shared__ float s_p[CHUNK];      // exp(s - m_c)
    __shared__ float s_red[128];
    __shared__ float s_acc[WAVES][DIM];

    const int b     = blockIdx.y;
    const int chunk = blockIdx.x;
    const int j0    = chunk * CHUNK;
    const int tid   = threadIdx.x;
    const int lane  = tid & 31;
    const int wave  = tid >> 5;

    if (tid < DIM) s_q[tid] = q[b * DIM + tid] * 0.125f;  // fold 1/temperature
    __syncthreads();

    const float* kb = k + (size_t)b * SEQ * DIM;
    const float* vb = v + (size_t)b * SEQ * DIM;

    // --- QK^T via WMMA f32 16x16x4 ---------------------------------------
    // A tile (16 rows x 4 K-cols): lane<16 -> K cols {k0,k0+1} of row lane,
    // lane>=16 -> K cols {k0+2,k0+3} of row (lane&15).
    const int mrow = lane & 15;
    const int ksel = (lane >> 4) << 1;   // 0 or 2

    for (int t = 0; t < TPW; ++t) {
        const int rowbase = j0 + wave * RPW + t * 16;
        const float* krow = kb + (size_t)(rowbase + mrow) * DIM + ksel;
        v8f c = {};
#pragma unroll
        for (int k0 = 0; k0 < DIM; k0 += 4) {
            v2f a  = *(const v2f*)(krow + k0);                 // global b64, coalesced
            v2f bq = { s_q[k0 + ksel], s_q[k0 + ksel + 1] };   // LDS broadcast
            c = __builtin_amdgcn_wmma_f32_16x16x4_f32(
                    /*neg_a=*/false, a, /*neg_b=*/false, bq,
                    /*c_mod=*/(short)0, c, /*reuse_a=*/false, /*reuse_b=*/false);
        }
        // Every column of D is identical (B columns replicated).
        // D layout: vgpr r = row r (lanes 0-15) / row 8+r (lanes 16-31).
        const int base = wave * RPW + t * 16;
        if (lane == 0) {
#pragma unroll
            for (int r = 0; r < 8; ++r) s_sc[base + r] = c[r];
        } else if (lane == 16) {
#pragma unroll
            for (int r = 0; r < 8; ++r) s_sc[base + 8 + r] = c[r];
        }
    }
    __syncthreads();

    // --- chunk-local max ---------------------------------------------------
    s_red[tid] = fmaxf(s_sc[tid], s_sc[tid + 128]);
    __syncthreads();
#pragma unroll
    for (int s = 64; s > 0; s >>= 1) {
        if (tid < s) s_red[tid] = fmaxf(s_red[tid], s_red[tid + s]);
        __syncthreads();
    }
    const float m_c = s_red[0];
    __syncthreads();

    // --- exp + chunk-local sum --------------------------------------------
    const float p0 = __expf(s_sc[tid]       - m_c);
    const float p1 = __expf(s_sc[tid + 128] - m_c);
    s_p[tid]       = p0;
    s_p[tid + 128] = p1;
    s_red[tid]     = p0 + p1;
    __syncthreads();
#pragma unroll
    for (int s = 64; s > 0; s >>= 1) {
        if (tid < s) s_red[tid] += s_red[tid + s];
        __syncthreads();
    }
    const float l_c = s_red[0];

    // --- partial P.V (fp32 FMA, coalesced b64 V loads) --------------------
    const int d0 = lane * 2;
    float acc0 = 0.f, acc1 = 0.f;
    const float* vrow = vb + (size_t)(j0 + wave * RPW) * DIM + d0;
#pragma unroll 8
    for (int jj = 0; jj < RPW; ++jj) {
        const float pj = s_p[wave * RPW + jj];
        v2f vv = *(const v2f*)(vrow + (size_t)jj * DIM);
        acc0 = fmaf(pj, vv.x, acc0);
        acc1 = fmaf(pj, vv.y, acc1);
    }
    s_acc[wave][d0]     = acc0;
    s_acc[wave][d0 + 1] = acc1;
    __syncthreads();

    // --- write partials + raw scores --------------------------------------
    float* wsb = ws + (size_t)(b * NCHUNK + chunk) * WSSTR;
    if (tid < DIM) {
        wsb[2 + tid] = s_acc[0][tid] + s_acc[1][tid] + s_acc[2][tid] + s_acc[3][tid];
    }
    if (tid == 0) { wsb[0] = m_c; wsb[1] = l_c; }

    attn_raw[(size_t)b * SEQ + j0 + tid]       = s_sc[tid];
    attn_raw[(size_t)b * SEQ + j0 + tid + 128] = s_sc[tid + 128];
}

// ---------------------------------------------------------------------------
// Pass 2: one block per batch. Reduce (m,l) across chunks, finalize out[64],
// normalize the attn row in place.
// ---------------------------------------------------------------------------
__global__ __launch_bounds__(256) void attn_pass2(
    const float* __restrict__ ws,
    float* __restrict__ out,    // d_out [B*64]
    float* __restrict__ attn)   // d_out + 4096: raw scores in, probs out
{
    __shared__ float s_m[NCHUNK];
    __shared__ float s_l[NCHUNK];
    __shared__ float s_ML[2];

    const int b   = blockIdx.x;
    const int tid = threadIdx.x;
    const float* wsb = ws + (size_t)b * NCHUNK * WSSTR;

    if (tid < NCHUNK) {
        s_m[tid] = wsb[(size_t)tid * WSSTR + 0];
        s_l[tid] = wsb[(size_t)tid * WSSTR + 1];
    }
    __syncthreads();
    if (tid == 0) {
        float M = -3.402823466e38f;
        for (int c = 0; c < NCHUNK; ++c) M = fmaxf(M, s_m[c]);
        float L = 0.f;
        for (int c = 0; c < NCHUNK; ++c) L += s_l[c] * __expf(s_m[c] - M);
        s_ML[0] = M;
        s_ML[1] = L;
    }
    __syncthreads();
    const float M    = s_ML[0];
    const float rcpL = 1.0f / s_ML[1];

    if (tid < DIM) {
        float a = 0.f;
        for (int c = 0; c < NCHUNK; ++c)
            a += wsb[(size_t)c * WSSTR + 2 + tid] * __expf(s_m[c] - M);
        out[b * DIM + tid] = a * rcpL;
    }

    for (int i = tid; i < SEQ; i += 256) {
        const float s = attn[(size_t)b * SEQ + i];
        attn[(size_t)b * SEQ + i] = __expf(s - M) * rcpL;
    }
}

// ---------------------------------------------------------------------------
extern "C" void kernel_launch(void* const* d_in, const int* in_sizes, int n_in,
                              void* d_out, int out_size, void* d_ws, size_t ws_size,
                              hipStream_t stream) {
    const float* q = (const float*)d_in[0];
    const float* k = (const float*)d_in[1];
    const float* v = (const float*)d_in[2];

    float* out  = (float*)d_out;          // [B,1,D] -> 4096 floats
    float* attn = out + BATCH * DIM;      // [B,1,S] -> 524288 floats
    float* ws   = (float*)d_ws;           // needs 64*32*68*4 = ~557 KB

    dim3 g1(NCHUNK, BATCH);
    attn_pass1<<<g1, 128, 0, stream>>>(q, k, v, attn, ws);
    attn_pass2<<<BATCH, 256, 0, stream>>>(ws, out, attn);
}